// VAE_81449759802044
// MI455X (gfx1250) — compile-verified
//
#include <hip/hip_runtime.h>

// CDNA5 / gfx1250, wave32. f32 WMMA 16x16x4 path.
typedef float v2f __attribute__((ext_vector_type(2)));
typedef float v8f __attribute__((ext_vector_type(8)));

#define MAXLEN 80
#define LATENT 10
#define NBATCH 256
#define NSTEPS 100
#define HID    64

static __device__ __forceinline__ v8f splat8(float s) {
    v8f v;
#pragma unroll
    for (int e = 0; e < 8; ++e) v[e] = s;
    return v;
}

// Counter-based Gaussian noise (hash -> Box-Muller). Deterministic per (ctr,id).
static __device__ __forceinline__ float gauss_noise(unsigned ctr, unsigned id) {
    unsigned x = ctr * 0x9E3779B1u + id * 0x85EBCA77u + 0x165667B1u;
    x ^= x >> 16; x *= 0x7FEB352Du; x ^= x >> 15; x *= 0x846CA68Bu; x ^= x >> 16;
    unsigned y = x * 0x9E3779B1u + 0x5851F42Du;
    y ^= y >> 16; y *= 0x7FEB352Du; y ^= y >> 15; y *= 0x846CA68Bu; y ^= y >> 16;
    float u1 = fmaxf((float)(x >> 8) * (1.0f / 16777216.0f), 1.0e-7f);
    float u2 = (float)(y >> 8) * (1.0f / 16777216.0f);
    return sqrtf(-2.0f * __logf(u1)) * __cosf(6.28318530718f * u2);
}

// One wave (32 threads) per block. blockIdx.x = sequence index i (0..79),
// blockIdx.y = batch tile (16 rows). D-layout convention (16x16 f32 C/D):
//   lane = n + 16*(m>=8); VGPR r holds row m = (lane>>4)*8 + r, col n = lane&15.
__global__ __launch_bounds__(32) void vae_diffusion_kernel(
    const float* __restrict__ zm,  const float* __restrict__ zlv,
    const float* __restrict__ W1,  const float* __restrict__ b1,
    const float* __restrict__ W2,  const float* __restrict__ b2,
    float* __restrict__ out)
{
    __shared__ float ldsXT[16 * 12];   // xt staged for A-operand reads (K padded to 12)
    __shared__ float hbuf[16 * 64];    // h staged for A-operand reads (K=64)

    const int  i    = blockIdx.x;
    const int  b0   = blockIdx.y * 16;
    const int  lane = threadIdx.x;
    const int  lid  = lane & 15;
    const int  half = lane >> 4;
    const bool valid = (lid < LATENT);

    // ---- per-element state in D-layout registers ----------------------------
    float xt[8], mu[8], asg[8], sstd[8], errv[8];
#pragma unroll
    for (int r = 0; r < 8; ++r) {
        int b   = b0 + half * 8 + r;
        int idx = b * (MAXLEN * LATENT) + i * LATENT + (valid ? lid : 0);
        float m  = valid ? zm[idx]  : 0.0f;
        float lv = valid ? zlv[idx] : 0.0f;
        float ez = __expf(lv);                 // reference: sigma tensor = exp(z_log_var)
        mu[r] = m;  xt[r] = m;
        asg[r]  = fabsf(ez);
        sstd[r] = __expf(0.5f * ez);           // sigma_std = exp(0.5*exp(z_log_var))
        errv[r] = 0.0f;
    }

    // ---- constant operand preloads ------------------------------------------
    float w1l[4];            // time row of W1 (row 800), per output column
    v8f   baseC[4];          // running pre-activation base (starts at b1)
#pragma unroll
    for (int j = 0; j < 4; ++j) {
        int c = 16 * j + lid;
        w1l[j]   = W1[800 * HID + c];
        baseC[j] = splat8(b1[c]);
    }

    // W1 rows for position i, B-operand layout. K padded 10->12; clamp rows <=800
    // (padded A columns are zero, so clamped B values never contribute).
    v2f w1b[3][4];
#pragma unroll
    for (int c3 = 0; c3 < 3; ++c3) {
#pragma unroll
        for (int j = 0; j < 4; ++j) {
            int k0 = i * LATENT + 4 * c3 + 2 * half;
            int r0 = (k0     < 800) ? k0     : 800;
            int r1 = (k0 + 1 < 800) ? k0 + 1 : 800;
            w1b[c3][j][0] = W1[r0 * HID + 16 * j + lid];
            w1b[c3][j][1] = W1[r1 * HID + 16 * j + lid];
        }
    }

    // W2 (64 x 10, pad N to 16 with zeros), B-operand layout, K=64.
    v2f w2b[16];
#pragma unroll
    for (int q = 0; q < 16; ++q) {
        int k0 = 4 * q + 2 * half;
        w2b[q][0] = valid ? W2[k0 * LATENT + lid]       : 0.0f;
        w2b[q][1] = valid ? W2[(k0 + 1) * LATENT + lid] : 0.0f;
    }
    const float b2v = valid ? b2[lid] : 0.0f;

    // ---- base_i = (prefix-masked mu) @ W1, K = 10*i, f32 WMMA 16x16x4 -------
    // Guard-free main loop over full K-chunks; single guarded tail (Ktot%4==2).
    const int Ktot  = i * LATENT;
    const int Kfull = Ktot & ~3;
    const float* zrow = zm + (b0 + lid) * (MAXLEN * LATENT);   // A row m = lid
    for (int kk = 0; kk < Kfull; kk += 4) {
        int ka = kk + 2 * half;                    // K split across lane halves
        v2f A = *(const v2f*)(zrow + ka);          // aligned b64 load
#pragma unroll
        for (int j = 0; j < 4; ++j) {
            v2f Bv;
            Bv[0] = W1[ka * HID + 16 * j + lid];
            Bv[1] = W1[(ka + 1) * HID + 16 * j + lid];
            baseC[j] = __builtin_amdgcn_wmma_f32_16x16x4_f32(
                false, A, false, Bv, (short)0, baseC[j], false, false);
        }
    }
    if (Kfull < Ktot) {                            // 2 leftover K values (half 0 only)
        int ka = Kfull + 2 * half;
        v2f A;
        if (half == 0) { A = *(const v2f*)(zrow + ka); }
        else           { A[0] = 0.0f; A[1] = 0.0f; }
        // B rows ka..ka+1 <= Ktot+1 <= 791 < 801: always in range; A==0 masks them.
#pragma unroll
        for (int j = 0; j < 4; ++j) {
            v2f Bv;
            Bv[0] = W1[ka * HID + 16 * j + lid];
            Bv[1] = W1[(ka + 1) * HID + 16 * j + lid];
            baseC[j] = __builtin_amdgcn_wmma_f32_16x16x4_f32(
                false, A, false, Bv, (short)0, baseC[j], false, false);
        }
    }

    // zero the two padded K columns of the xt staging buffer (written once)
    if (lid >= LATENT && lid < 12) {
#pragma unroll
        for (int r = 0; r < 8; ++r) ldsXT[(half * 8 + r) * 12 + lid] = 0.0f;
    }

    // ---- 100 sequential SDE steps -------------------------------------------
    const float fi = (float)i;
#pragma unroll 1
    for (int k = 0; k < NSTEPS; ++k) {
        float t = fi + 0.01f * (float)k;

        // xt: D-layout regs -> LDS (row-major 16x12) for A-operand reads
        if (valid) {
#pragma unroll
            for (int r = 0; r < 8; ++r)
                ldsXT[(half * 8 + r) * 12 + lid] = xt[r];
        }
        asm volatile("s_wait_dscnt 0" ::: "memory");

        v2f A3[3];
#pragma unroll
        for (int c3 = 0; c3 < 3; ++c3)   // even index pairs -> ds_load_b64
            A3[c3] = *(const v2f*)(&ldsXT[lid * 12 + 4 * c3 + 2 * half]);

        // pre = base + t*W1_timecol + xt@W1_i ; h = relu(pre)
        v8f h[4];
#pragma unroll
        for (int j = 0; j < 4; ++j) {
            v8f c = baseC[j] + splat8(t * w1l[j]);
#pragma unroll
            for (int c3 = 0; c3 < 3; ++c3)
                c = __builtin_amdgcn_wmma_f32_16x16x4_f32(
                    false, A3[c3], false, w1b[c3][j], (short)0, c, false, false);
#pragma unroll
            for (int e = 0; e < 8; ++e) c[e] = fmaxf(c[e], 0.0f);
            h[j] = c;
        }

        // h: D-layout -> LDS (row-major 16x64)
#pragma unroll
        for (int j = 0; j < 4; ++j) {
#pragma unroll
            for (int r = 0; r < 8; ++r)
                hbuf[(half * 8 + r) * 64 + 16 * j + lid] = h[j][r];
        }
        asm volatile("s_wait_dscnt 0" ::: "memory");

        // score = h @ W2 + b2, K=64 (16 K-chunks)
        v8f sc = splat8(b2v);
#pragma unroll
        for (int q = 0; q < 16; ++q) {   // even index pairs -> ds_load_b64
            v2f A = *(const v2f*)(&hbuf[lid * 64 + 4 * q + 2 * half]);
            sc = __builtin_amdgcn_wmma_f32_16x16x4_f32(
                false, A, false, w2b[q], (short)0, sc, false, false);
        }

        // elementwise: err accumulation + Euler-Maruyama update
        if (valid) {
#pragma unroll
            for (int r = 0; r < 8; ++r) {
                float s     = sc[r];
                float logdx = -(xt[r] - mu[r]) / asg[r];
                errv[r] += fabsf(logdx - s);
                int b = b0 + half * 8 + r;
                float n = gauss_noise((unsigned)(i * NSTEPS + k),
                                      (unsigned)(b * LATENT + lid));
                // reference: xt += 0.5*sigma_std^2*score + sigma_std*dW, dW = n*sqrt(dt)
                xt[r] += 0.5f * sstd[r] * sstd[r] * s + sstd[r] * 0.1f * n;
            }
        }
    }

    // ---- outputs: sequence then score_error, both [256,80,10] ---------------
    if (valid) {
#pragma unroll
        for (int r = 0; r < 8; ++r) {
            int  b = b0 + half * 8 + r;
            long o = (long)b * (MAXLEN * LATENT) + i * LATENT + lid;
            out[o] = xt[r];
            out[(long)NBATCH * MAXLEN * LATENT + o] = errv[r] * (1.0f / NSTEPS);
        }
    }
}

extern "C" void kernel_launch(void* const* d_in, const int* in_sizes, int n_in,
                              void* d_out, int out_size, void* d_ws, size_t ws_size,
                              hipStream_t stream) {
    (void)in_sizes; (void)n_in; (void)d_ws; (void)ws_size; (void)out_size;
    dim3 grid(MAXLEN, NBATCH / 16);   // 80 positions x 16 batch tiles, all independent
    vae_diffusion_kernel<<<grid, 32, 0, stream>>>(
        (const float*)d_in[0], (const float*)d_in[1], (const float*)d_in[2],
        (const float*)d_in[3], (const float*)d_in[4], (const float*)d_in[5],
        (float*)d_out);
}